// MTGNN_Att_2534030705017
// MI455X (gfx1250) — compile-verified
//
#include <hip/hip_runtime.h>
#include <math.h>

typedef __attribute__((ext_vector_type(2))) float v2f;
typedef __attribute__((ext_vector_type(4))) float v4f;
typedef __attribute__((ext_vector_type(8))) float v8f;
typedef __attribute__((ext_vector_type(4))) int   v4i;

#define NND   300
#define NP    304            // padded nodes (19*16)
#define DIMK  40
#define LFULL 168
#define LOUT  162
#define CCH   32
#define JB    (CCH*LOUT)     // 5184 columns per batch
#define NBATCH 16
#define KSEL  15
#define KT    8              // K-chunk rows staged per async step
#define NCH   (NP / KT)      // 38 chunks
#define BPITCH 520           // LDS B pitch (floats): +8 pad de-conflicts half-waves

// workspace offsets (floats)
#define OFF_NV1   0
#define OFF_NV2   12032
#define OFF_A     24064                 // 300*300 masked adjacency
#define OFF_AHAT  114176                // 304*304
#define OFF_AHATT 206720                // 304*304
#define OFF_WEFF  299264                // 2*(256 weights + 32 bias)
#define OFF_X2    300032                // 304*5184
#define OFF_H1    (OFF_X2 + NP*JB)
#define OFF_H2    (OFF_H1 + NP*JB)

__device__ __forceinline__ v8f wmma4(v2f a, v2f b, v8f c) {
  // V_WMMA_F32_16X16X4_F32 : D = A(16x4) x B(4x16) + C(16x16)
  return __builtin_amdgcn_wmma_f32_16x16x4_f32(false, a, false, b, (short)0, c,
                                               false, false);
}

// ---- gfx1250 async global->LDS staging (ASYNCcnt path), guarded fallback ----
#if __has_builtin(__builtin_amdgcn_global_load_async_to_lds_b128)
#define ASYNC_LDS 1
typedef __attribute__((address_space(1))) v4i GV4;   // global int4
typedef __attribute__((address_space(3))) v4i LV4;   // LDS int4
__device__ __forceinline__ void async_cp16(const float* g, float* l) {
  __builtin_amdgcn_global_load_async_to_lds_b128(
      (GV4*)(uintptr_t)g,
      (LV4*)(unsigned int)(uintptr_t)l, 0, 0);
}
#else
#define ASYNC_LDS 0
__device__ __forceinline__ void async_cp16(const float* g, float* l) {
  *(v4f*)l = *(const v4f*)g;   // synchronous fallback
}
#endif

template <int N> __device__ __forceinline__ void wait_async() {
#if ASYNC_LDS
#if __has_builtin(__builtin_amdgcn_s_wait_asynccnt)
  __builtin_amdgcn_s_wait_asynccnt((short)N);
#else
  asm volatile("s_wait_asynccnt %0" ::"n"(N) : "memory");
#endif
#endif
}

// ---------------- graph construction ----------------
__global__ void nodevec_kernel(const float* __restrict__ emb1, const float* __restrict__ emb2,
                               const float* __restrict__ W1, const float* __restrict__ b1,
                               const float* __restrict__ W2, const float* __restrict__ b2,
                               float* __restrict__ nv1, float* __restrict__ nv2) {
  int g = blockIdx.x * 256 + threadIdx.x;
  if (g >= 2 * NND * DIMK) return;
  int which = g / (NND * DIMK);
  int rem = g - which * NND * DIMK;
  int i = rem / DIMK, o = rem - (rem / DIMK) * DIMK;
  const float* emb = which ? emb2 : emb1;
  const float* W   = which ? W2 : W1;
  const float* bb  = which ? b2 : b1;
  float s = bb[o];
  for (int d = 0; d < DIMK; ++d) s += emb[i * DIMK + d] * W[o * DIMK + d];
  (which ? nv2 : nv1)[i * DIMK + o] = tanhf(3.0f * s);
}

__global__ void adj_kernel(const float* __restrict__ nv1, const float* __restrict__ nv2,
                           float* __restrict__ A) {
  int g = blockIdx.x * 256 + threadIdx.x;
  if (g >= NND * NND) return;
  int i = g / NND, j = g - i * NND;
  float s = 0.f;
  for (int d = 0; d < DIMK; ++d)
    s += nv1[i * DIMK + d] * nv2[j * DIMK + d] - nv2[i * DIMK + d] * nv1[j * DIMK + d];
  float a = tanhf(3.0f * s);
  A[g] = a > 0.f ? a : 0.f;
}

__global__ __launch_bounds__(32) void topk_kernel(float* __restrict__ A) {
  __shared__ float vals[NND];
  __shared__ int   sel[NND];
  __shared__ float rv[32];
  __shared__ int   ri[32];
  int row = blockIdx.x, lane = threadIdx.x;
  for (int j = lane; j < NND; j += 32) { vals[j] = A[row * NND + j]; sel[j] = 0; }
  __syncthreads();
  for (int it = 0; it < KSEL; ++it) {
    float bv = -1e30f; int bi = 0x7fffffff;
    for (int j = lane; j < NND; j += 32)
      if (!sel[j]) { float x = vals[j]; if (x > bv) { bv = x; bi = j; } }
    rv[lane] = bv; ri[lane] = bi;
    __syncthreads();
    if (lane == 0) {
      float Bv = rv[0]; int Bi = ri[0];
      for (int t = 1; t < 32; ++t)
        if (rv[t] > Bv || (rv[t] == Bv && ri[t] < Bi)) { Bv = rv[t]; Bi = ri[t]; }
      if (Bi < NND) sel[Bi] = 1;
    }
    __syncthreads();
  }
  for (int j = lane; j < NND; j += 32) A[row * NND + j] = sel[j] ? vals[j] : 0.f;
}

__global__ __launch_bounds__(128) void norm_kernel(const float* __restrict__ Am,
                                                   float* __restrict__ Ahat,
                                                   float* __restrict__ AhatT) {
  __shared__ float redf[128];
  __shared__ float redb[128];
  int i = blockIdx.x, t = threadIdx.x;
  if (i >= NND) {
    for (int j = t; j < NP; j += 128) { Ahat[i * NP + j] = 0.f; AhatT[i * NP + j] = 0.f; }
    return;
  }
  float sf = 0.f, sb = 0.f;
  for (int j = t; j < NND; j += 128) { sf += Am[i * NND + j]; sb += Am[j * NND + i]; }
  redf[t] = sf; redb[t] = sb;
  __syncthreads();
  for (int s = 64; s > 0; s >>= 1) {
    if (t < s) { redf[t] += redf[t + s]; redb[t] += redb[t + s]; }
    __syncthreads();
  }
  float rsf = 1.f + redf[0];
  float rsb = 1.f + redb[0];
  for (int j = t; j < NP; j += 128) {
    if (j < NND) {
      float af = Am[i * NND + j], ab = Am[j * NND + i];
      if (j == i) { af += 1.f; ab += 1.f; }
      Ahat[i * NP + j]  = af / rsf;
      AhatT[i * NP + j] = ab / rsb;
    } else {
      Ahat[i * NP + j] = 0.f; AhatT[i * NP + j] = 0.f;
    }
  }
}

// ------- collapse start-conv (CIN=1) into inception weights -------
__global__ __launch_bounds__(64) void weff_kernel(
    const float* __restrict__ Ws, const float* __restrict__ bs,
    const float* fw2, const float* fb2, const float* fw3, const float* fb3,
    const float* fw6, const float* fb6, const float* fw7, const float* fb7,
    const float* gw2, const float* gb2, const float* gw3, const float* gb3,
    const float* gw6, const float* gb6, const float* gw7, const float* gb7,
    float* __restrict__ weff) {
  int t = threadIdx.x;
  if (t >= 64) return;
  int fg = t >> 5, cg = t & 31, br = cg >> 3, o = cg & 7;
  int kern = (br == 0) ? 2 : ((br == 1) ? 3 : ((br == 2) ? 6 : 7));
  const float* wp; const float* bp;
  switch (fg * 4 + br) {
    case 0: wp = fw2; bp = fb2; break;  case 1: wp = fw3; bp = fb3; break;
    case 2: wp = fw6; bp = fb6; break;  case 3: wp = fw7; bp = fb7; break;
    case 4: wp = gw2; bp = gb2; break;  case 5: wp = gw3; bp = gb3; break;
    case 6: wp = gw6; bp = gb6; break;  default: wp = gw7; bp = gb7; break;
  }
  for (int j = 0; j < 8; ++j) {
    float s = 0.f;
    if (j < kern)
      for (int c = 0; c < CCH; ++c) s += wp[(o * CCH + c) * kern + j] * Ws[c];
    weff[fg * 288 + cg * 8 + j] = s;
  }
  float be = bp[o];
  for (int c = 0; c < CCH; ++c) {
    float acc = 0.f;
    for (int j = 0; j < kern; ++j) acc += wp[(o * CCH + c) * kern + j];
    be += acc * bs[c];
  }
  weff[fg * 288 + 256 + cg] = be;
}

// ------- inception (collapsed) + tanh*sigmoid gate, per batch -------
__global__ void inception_kernel(const float* __restrict__ X,
                                 const float* __restrict__ weff,
                                 float* __restrict__ X2b, int n) {
  int g = blockIdx.x * 256 + threadIdx.x;
  if (g >= NP * JB) return;
  int v = g / JB;
  int j = g - v * JB;
  if (v >= NND) { X2b[g] = 0.f; return; }      // zero node padding rows
  int c = j / LOUT, l = j - c * LOUT;
  int br = c >> 3;
  int kern = (br == 0) ? 2 : ((br == 1) ? 3 : ((br == 2) ? 6 : 7));
  const float* xp = X + ((long)n * NND + v) * LFULL + (7 - kern + l);
  float xf = weff[256 + c];
  float xg = weff[288 + 256 + c];
  for (int jj = 0; jj < kern; ++jj) {
    float t = xp[jj];
    xf += weff[c * 8 + jj] * t;
    xg += weff[288 + c * 8 + jj] * t;
  }
  float sg = 1.0f / (1.0f + __expf(-xg));
  X2b[g] = tanhf(xf) * sg;
}

// ------- mixprop node-mixing GEMM: Hout = 0.05*X2 + 0.95*(Ahat x Hin) -------
// A panel (16x304) in LDS; B panel async-DMA'd to LDS in KT-row double buffers.
__global__ __launch_bounds__(256) void mixprop_kernel(const float* __restrict__ Ahat,
                                                      const float* __restrict__ Hin,
                                                      const float* __restrict__ Xb,
                                                      float* __restrict__ Hout) {
  __shared__ float As[16 * 308];               // pitch 308 -> conflict-free frags
  __shared__ float Bs[2][KT * BPITCH];
  int row0 = blockIdx.y * 16;
  int tid = threadIdx.x;
  for (int t = tid; t < 16 * 304; t += 256) {
    int r = t / 304, cc = t - r * 304;
    As[r * 308 + cc] = Ahat[(row0 + r) * NP + cc];
  }
  int col0b = blockIdx.x * 512;

  // stage KT x 512 floats: 1024 b128 slots, 4 per thread (4 async instrs/wave)
  auto issue = [&](int kbase, int buf) {
    for (int i = 0; i < 4; ++i) {
      int s = tid + i * 256;
      int r = s >> 7;                 // 0..KT-1
      int cq = (s & 127) << 2;        // 0..508 step 4
      int gc = col0b + cq;
      if (gc >= JB) gc = JB - 4;      // clamp tail block (cols >= JB unused)
      async_cp16(Hin + (long)(kbase + r) * JB + gc, &Bs[buf][r * BPITCH + cq]);
    }
  };

  issue(0, 0);
  __syncthreads();                    // As staged (async chunk0 still in flight)

  int wave = tid >> 5;
  int lane = tid & 31;
  int h2 = (lane >> 4) << 1;          // K offset 0 / 2 per half-wave
  int ln = lane & 15;
  int colw = wave * 64;
  v8f acc0 = {}, acc1 = {}, acc2 = {}, acc3 = {};
  int cur = 0;
  for (int kc = 0; kc < NCH; ++kc) {
    if (kc + 1 < NCH) { issue((kc + 1) * KT, cur ^ 1); wait_async<4>(); }
    else              { wait_async<0>(); }
    __syncthreads();                  // chunk kc visible to all waves
#pragma unroll
    for (int kk = 0; kk < KT; kk += 4) {
      int k = kc * KT + kk;
      v2f af = *(const v2f*)&As[ln * 308 + k + h2];
      const float* bp0 = &Bs[cur][(kk + h2) * BPITCH + colw + ln];
      const float* bp1 = bp0 + BPITCH;
      v2f b0; b0.x = bp0[0];  b0.y = bp1[0];
      v2f b1; b1.x = bp0[16]; b1.y = bp1[16];
      v2f b2; b2.x = bp0[32]; b2.y = bp1[32];
      v2f b3; b3.x = bp0[48]; b3.y = bp1[48];
      acc0 = wmma4(af, b0, acc0);
      acc1 = wmma4(af, b1, acc1);
      acc2 = wmma4(af, b2, acc2);
      acc3 = wmma4(af, b3, acc3);
    }
    __syncthreads();                  // done reading Bs[cur] before refill
    cur ^= 1;
  }
  int col = col0b + colw + ln;
  for (int r = 0; r < 8; ++r) {
    int row = row0 + r + (h2 << 2);
    long base = (long)row * JB + col;
    if (col      < JB) Hout[base     ] = 0.05f * Xb[base     ] + 0.95f * acc0[r];
    if (col + 16 < JB) Hout[base + 16] = 0.05f * Xb[base + 16] + 0.95f * acc1[r];
    if (col + 32 < JB) Hout[base + 32] = 0.05f * Xb[base + 32] + 0.95f * acc2[r];
    if (col + 48 < JB) Hout[base + 48] = 0.05f * Xb[base + 48] + 0.95f * acc3[r];
  }
}

// ------- output 1x1 conv over concat(X2,H1,H2): WMMA GEMM M=32,K=96 -------
__global__ __launch_bounds__(256) void conv_kernel(const float* __restrict__ Wm,
                                                   const float* __restrict__ bm1,
                                                   const float* __restrict__ bm2,
                                                   const float* __restrict__ Xb,
                                                   const float* __restrict__ H1,
                                                   const float* __restrict__ H2,
                                                   float* __restrict__ out,
                                                   int n, int accflag) {
  int wid = blockIdx.x * 8 + (threadIdx.x >> 5);
  if (wid >= NND * 11) return;
  int v = wid / 11, lt = wid - v * 11, l0 = lt * 16;
  int lane = threadIdx.x & 31;
  int h2 = (lane >> 4) << 1;
  int ln = lane & 15;
  v8f acc0 = {}, acc1 = {};
  for (int k = 0; k < 96; k += 4) {
    int r0 = k + h2;                            // K row pair base
    const float* hbuf = (r0 < 32) ? Xb : ((r0 < 64) ? H1 : H2);
    int c0 = r0 & 31;
    const float* p = hbuf + (long)v * JB + c0 * LOUT + l0 + ln;
    v2f b; b.x = p[0]; b.y = p[LOUT];
    v2f a0 = *(const v2f*)(Wm + ln * 96 + r0);
    v2f a1 = *(const v2f*)(Wm + (16 + ln) * 96 + r0);
    acc0 = wmma4(a0, b, acc0);
    acc1 = wmma4(a1, b, acc1);
  }
  int l = l0 + ln;
  if (l >= LOUT) return;
  for (int r = 0; r < 8; ++r) {
    int o0 = r + (h2 << 2);
    int o1 = o0 + 16;
    long i0 = (((long)n * CCH + o0) * NND + v) * LOUT + l;
    long i1 = (((long)n * CCH + o1) * NND + v) * LOUT + l;
    float v0 = acc0[r], v1 = acc1[r];
    if (accflag) { v0 += out[i0]; v1 += out[i1]; }
    else         { v0 += bm1[o0] + bm2[o0]; v1 += bm1[o1] + bm2[o1]; }
    out[i0] = v0; out[i1] = v1;
  }
}

extern "C" void kernel_launch(void* const* d_in, const int* in_sizes, int n_in,
                              void* d_out, int out_size, void* d_ws, size_t ws_size,
                              hipStream_t stream) {
  const float* X    = (const float*)d_in[1];
  const float* emb1 = (const float*)d_in[2];
  const float* emb2 = (const float*)d_in[3];
  const float* W1   = (const float*)d_in[4];
  const float* b1   = (const float*)d_in[5];
  const float* W2   = (const float*)d_in[6];
  const float* b2   = (const float*)d_in[7];
  const float* Ws   = (const float*)d_in[8];
  const float* bsv  = (const float*)d_in[9];
  const float* fw2 = (const float*)d_in[10]; const float* fb2 = (const float*)d_in[11];
  const float* fw3 = (const float*)d_in[12]; const float* fb3 = (const float*)d_in[13];
  const float* fw6 = (const float*)d_in[14]; const float* fb6 = (const float*)d_in[15];
  const float* fw7 = (const float*)d_in[16]; const float* fb7 = (const float*)d_in[17];
  const float* gw2 = (const float*)d_in[18]; const float* gb2 = (const float*)d_in[19];
  const float* gw3 = (const float*)d_in[20]; const float* gb3 = (const float*)d_in[21];
  const float* gw6 = (const float*)d_in[22]; const float* gb6 = (const float*)d_in[23];
  const float* gw7 = (const float*)d_in[24]; const float* gb7 = (const float*)d_in[25];
  const float* Wm1 = (const float*)d_in[26]; const float* bm1 = (const float*)d_in[27];
  const float* Wm2 = (const float*)d_in[28]; const float* bm2 = (const float*)d_in[29];

  float* ws    = (float*)d_ws;
  float* nv1   = ws + OFF_NV1;
  float* nv2   = ws + OFF_NV2;
  float* Am    = ws + OFF_A;
  float* Ahat  = ws + OFF_AHAT;
  float* AhatT = ws + OFF_AHATT;
  float* weff  = ws + OFF_WEFF;
  float* X2b   = ws + OFF_X2;
  float* H1    = ws + OFF_H1;
  float* H2    = ws + OFF_H2;
  float* out   = (float*)d_out;

  nodevec_kernel<<<(2 * NND * DIMK + 255) / 256, 256, 0, stream>>>(emb1, emb2, W1, b1, W2, b2, nv1, nv2);
  adj_kernel<<<(NND * NND + 255) / 256, 256, 0, stream>>>(nv1, nv2, Am);
  topk_kernel<<<NND, 32, 0, stream>>>(Am);
  norm_kernel<<<NP, 128, 0, stream>>>(Am, Ahat, AhatT);
  weff_kernel<<<1, 64, 0, stream>>>(Ws, bsv, fw2, fb2, fw3, fb3, fw6, fb6, fw7, fb7,
                                    gw2, gb2, gw3, gb3, gw6, gb6, gw7, gb7, weff);

  dim3 mpGrid(11, 19);   // 11 blocks x 512 cols cover 5184; 19 row-tiles
  int convBlocks = (NND * 11 + 7) / 8;
  for (int n = 0; n < NBATCH; ++n) {
    inception_kernel<<<(NP * JB) / 256, 256, 0, stream>>>(X, weff, X2b, n);
    // forward mixprop
    mixprop_kernel<<<mpGrid, 256, 0, stream>>>(Ahat, X2b, X2b, H1);
    mixprop_kernel<<<mpGrid, 256, 0, stream>>>(Ahat, H1, X2b, H2);
    conv_kernel<<<convBlocks, 256, 0, stream>>>(Wm1, bm1, bm2, X2b, H1, H2, out, n, 0);
    // backward (transposed-adjacency) mixprop, reusing H buffers
    mixprop_kernel<<<mpGrid, 256, 0, stream>>>(AhatT, X2b, X2b, H1);
    mixprop_kernel<<<mpGrid, 256, 0, stream>>>(AhatT, H1, X2b, H2);
    conv_kernel<<<convBlocks, 256, 0, stream>>>(Wm2, bm1, bm2, X2b, H1, H2, out, n, 1);
  }
}